// Nonlinear_910533067374
// MI455X (gfx1250) — compile-verified
//
#include <hip/hip_runtime.h>

// Nonlinear + pairwise-product fused streaming kernel for MI455X (gfx1250).
//
// Input : x  (nrows, 128) fp32, row = [64 unary-cols | 64 product-cols]
// Output: out(nrows,  96) fp32, row = [id/sin/cos/sigmoid pattern | 32 products]
//
// Memory-bound (~0.94 GB @ 23.3 TB/s). Strategy: b128 non-temporal loads/stores,
// native v_sin/v_cos/v_exp/v_rcp, period-4 pattern aligned to float4 so there is
// no per-lane divergence inside the unary path.

typedef __attribute__((ext_vector_type(4))) float v4f;

__global__ __launch_bounds__(256) void nonlinear_prod_kernel(
    const float* __restrict__ x, float* __restrict__ out, unsigned nrows) {
  const unsigned tid   = blockIdx.x * blockDim.x + threadIdx.x;
  const unsigned total = nrows * 24u;            // 24 output float4s per row
  if (tid >= total) return;

  const unsigned row = tid / 24u;                // const-div -> mulhi+shift
  const unsigned c   = tid - row * 24u;

  const v4f* __restrict__ xrow = reinterpret_cast<const v4f*>(x) + (size_t)row * 32u;
  v4f*       __restrict__ orow = reinterpret_cast<v4f*>(out)     + (size_t)row * 24u;

  if (c < 16u) {
    // Unary part: float4 covers cols {4c, 4c+1, 4c+2, 4c+3} -> {id, sin, cos, sigmoid}
    v4f v = __builtin_nontemporal_load(xrow + c);
    v4f r;
    r.x = v.x;                                   // identity
    r.y = __sinf(v.y);                           // v_sin_f32 (scaled by 1/2pi in header)
    r.z = __cosf(v.z);                           // v_cos_f32
    // sigmoid = 1 / (1 + exp(-x)) via native exp2 + rcp
    {
      float e = __builtin_amdgcn_exp2f(-v.w * 1.44269504088896340736f); // v_exp_f32
      r.w = __builtin_amdgcn_rcpf(1.0f + e);                            // v_rcp_f32
    }
    __builtin_nontemporal_store(r, orow + c);
  } else {
    // Product part: out float4 c covers prod[4m..4m+3], m = c-16,
    // sourced from input float4s (16+2m) and (16+2m+1) of this row.
    const unsigned m = c - 16u;
    v4f a = __builtin_nontemporal_load(xrow + 16u + 2u * m);
    v4f b = __builtin_nontemporal_load(xrow + 17u + 2u * m);
    v4f r;
    r.x = a.x * a.y;
    r.y = a.z * a.w;
    r.z = b.x * b.y;
    r.w = b.z * b.w;
    __builtin_nontemporal_store(r, orow + c);
  }
}

extern "C" void kernel_launch(void* const* d_in, const int* in_sizes, int n_in,
                              void* d_out, int out_size, void* d_ws, size_t ws_size,
                              hipStream_t stream) {
  (void)n_in; (void)out_size; (void)d_ws; (void)ws_size;
  const float* x = (const float*)d_in[0];
  float* out = (float*)d_out;

  const unsigned row_elems = 128u;               // U + 2V
  const unsigned nrows = (unsigned)(in_sizes[0] / (int)row_elems);
  const unsigned total = nrows * 24u;            // output float4s

  const unsigned block = 256u;                   // 8 wave32s per block
  const unsigned grid  = (total + block - 1u) / block;
  nonlinear_prod_kernel<<<dim3(grid), dim3(block), 0, stream>>>(x, out, nrows);
}